// HardTripletLoss_65395172049023
// MI455X (gfx1250) — compile-verified
//
#include <hip/hip_runtime.h>
#include <hip/hip_bf16.h>
#include <math.h>

typedef __attribute__((ext_vector_type(2))) float v2f;
typedef __attribute__((ext_vector_type(8))) float v8f;

#define N_   2
#define C_   64
#define HR_  60
#define WR_  80
#define K_   1024
#define HW_  (HR_ * WR_)      // 4800
#define GSF  8

// workspace layout (float offsets)
#define OFF_A2   0
#define OFF_KGX  (OFF_A2  + K_)
#define OFF_KGY  (OFF_KGX + K_)
#define OFF_WGX  (OFF_KGY + K_)
#define OFF_WGY  (OFF_WGX + N_*HW_)
#define OFF_VM   (OFF_WGY + N_*HW_)
#define OFF_B2   (OFF_VM  + N_*HW_)
#define OFF_POS  (OFF_B2  + N_*HW_)
#define OFF_NEG  (OFF_POS + N_*K_)

// -------- Kernel 1: bilinear descriptor sampling + ||a||^2 + kp_grid --------
__global__ void __launch_bounds__(C_) sample_kernel(
    const float* __restrict__ kp1, const float* __restrict__ desc1,
    float* __restrict__ out_desc, float* __restrict__ ws) {
  const int k = blockIdx.x;
  const int c = threadIdx.x;
  const float bf = kp1[k * 4 + 0];
  const float yy = kp1[k * 4 + 2];
  const float xx = kp1[k * 4 + 3];
  const int bi = (int)bf;
  float xg = fminf(fmaxf(xx * 0.125f, 0.0f), (float)(WR_ - 1));
  float yg = fminf(fmaxf(yy * 0.125f, 0.0f), (float)(HR_ - 1));
  float x0 = floorf(xg), y0 = floorf(yg);
  float x1 = fminf(x0 + 1.0f, (float)(WR_ - 1));
  float y1 = fminf(y0 + 1.0f, (float)(HR_ - 1));
  float wx = xg - x0, wy = yg - y0;
  int x0i = (int)x0, x1i = (int)x1, y0i = (int)y0, y1i = (int)y1;
  const float* base = desc1 + ((size_t)bi * C_ + c) * HW_;
  float v00 = base[y0i * WR_ + x0i];
  float v01 = base[y0i * WR_ + x1i];
  float v10 = base[y1i * WR_ + x0i];
  float v11 = base[y1i * WR_ + x1i];
  float val = (1.0f - wy) * ((1.0f - wx) * v00 + wx * v01)
            +         wy  * ((1.0f - wx) * v10 + wx * v11);
  out_desc[k * C_ + c] = val;
  __shared__ float s[C_];
  s[c] = val * val;
  __syncthreads();
  for (int off = C_ / 2; off > 0; off >>= 1) {
    if (c < off) s[c] += s[c + off];
    __syncthreads();
  }
  if (c == 0) {
    ws[OFF_A2 + k] = s[0];
    // replicate reference quirk: only keypoints 0 and 1 are divided by GS
    float sc = (k < 2) ? 0.125f : 1.0f;
    ws[OFF_KGX + k] = xx * sc;
    ws[OFF_KGY + k] = yy * sc;
  }
}

// -------- Kernel 2: warped grid, visibility product, ||b||^2 --------
__global__ void prep_kernel(const float* __restrict__ desc2,
                            const float* __restrict__ homo21,
                            const float* __restrict__ vis,
                            float* __restrict__ ws) {
  int idx = blockIdx.x * blockDim.x + threadIdx.x;
  if (idx >= N_ * HW_) return;
  int n  = idx / HW_;
  int hw = idx - n * HW_;
  int h  = hw / WR_;
  int w  = hw - h * WR_;
  float xf = (float)(w * GSF + GSF / 2);
  float yf = (float)(h * GSF + GSF / 2);
  const float* H = homo21 + n * 9;
  float w0 = H[0] * xf + H[1] * yf + H[2];
  float w1 = H[3] * xf + H[4] * yf + H[5];
  float w2 = H[6] * xf + H[7] * yf + H[8];
  ws[OFF_WGX + idx] = w0 / w2;
  ws[OFF_WGY + idx] = w1 / w2;
  const float* vb = vis + (size_t)n * (HR_ * GSF) * (WR_ * GSF);
  float p = 1.0f;
  for (int i = 0; i < GSF; i++)
    for (int j = 0; j < GSF; j++)
      p *= vb[(h * GSF + i) * (WR_ * GSF) + (w * GSF + j)];
  ws[OFF_VM + idx] = p;
  const float* d2 = desc2 + (size_t)n * C_ * HW_ + hw;
  float sacc = 0.0f;
  for (int c = 0; c < C_; c++) { float v = d2[c * HW_]; sacc += v * v; }
  ws[OFF_B2 + idx] = sacc;
}

// -------- Kernel 3: fused WMMA-f32 GEMM + dist + masked max/min reduction --------
__global__ void __launch_bounds__(256) gemm_reduce_kernel(
    const float* __restrict__ adesc,   // kp1_desc, K x C row-major
    const float* __restrict__ desc2,
    float* __restrict__ ws) {
  const int n     = blockIdx.x >> 6;          // 2 * 64 blocks
  const int mbase = (blockIdx.x & 63) * 16;   // k-row tile base
  const int tid   = threadIdx.x;
  const int wv    = tid >> 5;                 // 8 waves
  const int lane  = tid & 31;
  const int half  = lane >> 4;
  const int l15   = lane & 15;
  const int koff  = half * 2;                 // A/B K-split: lanes 16-31 hold K+2

  // preload A tile rows per WMMA 16x16x4 f32 layout
  const int mA = mbase + l15;
  float ax[16], ay[16];
  #pragma unroll
  for (int kk = 0; kk < 16; kk++) {
    ax[kk] = adesc[mA * C_ + kk * 4 + koff];
    ay[kk] = adesc[mA * C_ + kk * 4 + koff + 1];
  }
  // per-lane D-row metadata: VGPR r holds rows mbase + half*8 + r
  float a2v[8], kgx[8], kgy[8];
  #pragma unroll
  for (int r = 0; r < 8; r++) {
    int m = mbase + half * 8 + r;
    a2v[r] = ws[OFF_A2  + m];
    kgx[r] = ws[OFF_KGX + m];
    kgy[r] = ws[OFF_KGY + m];
  }
  const float* B   = desc2 + (size_t)n * C_ * HW_;
  const float* wgx = ws + OFF_WGX + n * HW_;
  const float* wgy = ws + OFF_WGY + n * HW_;
  const float* vmp = ws + OFF_VM  + n * HW_;
  const float* b2p = ws + OFF_B2  + n * HW_;

  float pmax[8], nmin[8];
  #pragma unroll
  for (int r = 0; r < 8; r++) { pmax[r] = -INFINITY; nmin[r] = INFINITY; }

  // waves stripe over the 300 column tiles (trip counts wave-uniform: EXEC all-ones)
  for (int t = wv; t < HW_ / 16; t += 8) {
    const int hw = t * 16 + l15;
    v8f acc = {0.f, 0.f, 0.f, 0.f, 0.f, 0.f, 0.f, 0.f};
    #pragma unroll
    for (int kk = 0; kk < 16; kk++) {
      v2f a; a[0] = ax[kk]; a[1] = ay[kk];
      v2f b;
      b[0] = B[(kk * 4 + koff)     * HW_ + hw];
      b[1] = B[(kk * 4 + koff + 1) * HW_ + hw];
      acc = __builtin_amdgcn_wmma_f32_16x16x4_f32(
          false, a, false, b, (short)0, acc, false, false);
    }
    const float wxv = wgx[hw], wyv = wgy[hw], v = vmp[hw], b2 = b2p[hw];
    #pragma unroll
    for (int r = 0; r < 8; r++) {
      float d2   = a2v[r] + b2 - 2.0f * acc[r];
      float dist = sqrtf(fmaxf(d2, 1e-12f));
      float dx = kgx[r] - wxv, dy = kgy[r] - wyv;
      float gd = sqrtf(dx * dx + dy * dy);
      float sr = (gd <= (float)GSF - 0.5f) ? 1.0f : 0.0f;
      float pc = dist - (1.0f - sr) * v * 5.0f;   // dist - ns*5
      float nc = dist +         sr  * v * 5.0f;   // dist + s*5
      pmax[r] = fmaxf(pmax[r], pc);
      nmin[r] = fminf(nmin[r], nc);
    }
  }
  // reduce over the 16 column lanes (each 16-lane half holds distinct rows)
  #pragma unroll
  for (int r = 0; r < 8; r++) {
    #pragma unroll
    for (int off = 1; off < 16; off <<= 1) {
      pmax[r] = fmaxf(pmax[r], __shfl_xor(pmax[r], off, 16));
      nmin[r] = fminf(nmin[r], __shfl_xor(nmin[r], off, 16));
    }
  }
  __shared__ float lp[8][16], ln[8][16];
  if (l15 == 0) {
    #pragma unroll
    for (int r = 0; r < 8; r++) {
      lp[wv][half * 8 + r] = pmax[r];
      ln[wv][half * 8 + r] = nmin[r];
    }
  }
  __syncthreads();
  if (tid < 16) {
    float p = -INFINITY, q = INFINITY;
    for (int w = 0; w < 8; w++) { p = fmaxf(p, lp[w][tid]); q = fminf(q, ln[w][tid]); }
    ws[OFF_POS + n * K_ + mbase + tid] = p;
    ws[OFF_NEG + n * K_ + mbase + tid] = q;
  }
}

// -------- Kernel 4: scalar triplet loss --------
__global__ void __launch_bounds__(256) loss_kernel(const float* __restrict__ ws,
                                                   float* __restrict__ out) {
  __shared__ float s[256];
  int tid = threadIdx.x;
  float acc = 0.0f;
  for (int i = tid; i < N_ * K_; i += 256) {
    float v = ws[OFF_POS + i] - ws[OFF_NEG + i] + 1.0f;  // MARGIN = 1
    acc += fmaxf(v, 0.0f);
  }
  s[tid] = acc;
  __syncthreads();
  for (int off = 128; off > 0; off >>= 1) {
    if (tid < off) s[tid] += s[tid + off];
    __syncthreads();
  }
  if (tid == 0) out[0] = s[0] / (float)(N_ * K_);  // LOSS_LAMBDA = 1
}

extern "C" void kernel_launch(void* const* d_in, const int* in_sizes, int n_in,
                              void* d_out, int out_size, void* d_ws, size_t ws_size,
                              hipStream_t stream) {
  const float* kp1    = (const float*)d_in[0];
  const float* desc1  = (const float*)d_in[1];
  const float* desc2  = (const float*)d_in[2];
  const float* homo21 = (const float*)d_in[3];
  const float* vis    = (const float*)d_in[4];
  float* out = (float*)d_out;          // [0] = loss, [1..] = kp1_desc (K x C)
  float* ws  = (float*)d_ws;

  sample_kernel<<<K_, C_, 0, stream>>>(kp1, desc1, out + 1, ws);
  prep_kernel<<<(N_ * HW_ + 255) / 256, 256, 0, stream>>>(desc2, homo21, vis, ws);
  gemm_reduce_kernel<<<N_ * (K_ / 16), 256, 0, stream>>>(out + 1, desc2, ws);
  loss_kernel<<<1, 256, 0, stream>>>(ws, out);
}